// EMDLoss_63178968924668
// MI455X (gfx1250) — compile-verified
//
#include <hip/hip_runtime.h>
#include <hip/hip_bf16.h>
#include <stdint.h>

// Shapes fixed by the reference setup
#define N_ 32
#define C_ 256
#define S_ 4096
#define TS 256           // s-columns per block (== blockDim.x, 8 waves)
#define CH 8             // c-rows staged per chunk
#define NCHUNK (C_/CH)   // 32 chunks
#define STILES (S_/TS)   // 16 s-tiles per n
#define NBLK (N_*STILES) // 512 blocks -> 512 partials

typedef __attribute__((ext_vector_type(2))) float v2f;
typedef __attribute__((ext_vector_type(8))) float v8f;

// ---------------------------------------------------------------------------
// Main kernel: per-column cumsum/L2 scan, fed by double-buffered async copies
// (global_load_async_to_lds_b128 + s_wait_asynccnt, CDNA5 async path).
// ---------------------------------------------------------------------------
__global__ __launch_bounds__(TS) void emd_main(const float* __restrict__ inp,
                                               const int* __restrict__ tgt,
                                               float* __restrict__ partial) {
  __shared__ __align__(16) float lbuf[2][CH * TS];

  const int tid = threadIdx.x;
  const int bid = blockIdx.x;
  const int n   = bid / STILES;
  const int s0  = (bid % STILES) * TS;

  // Producer addressing: 256 threads move CH*TS floats (8 KB) per chunk as
  // 2 x b128 per thread. row = tid>>6 in 0..3 (and row+4), col multiple of 4.
  const unsigned row     = (unsigned)tid >> 6;
  const unsigned col     = ((unsigned)tid & 63u) << 2;          // float index
  const unsigned voff0   = (row * S_ + col) * 4u;               // global byte off
  const unsigned voff1   = voff0 + 4u * S_ * 4u;
  const unsigned ldsA    = (unsigned)(uintptr_t)&lbuf[0][row * TS + col];
  const unsigned ldsB    = (unsigned)(uintptr_t)&lbuf[1][row * TS + col];
  const unsigned ldsStep = 4u * TS * 4u;                        // 4 rows in bytes

  const float* base = inp + ((size_t)n * C_) * S_ + s0;

  // Prologue: chunk 0 -> buffer 0
  asm volatile("global_load_async_to_lds_b128 %0, %1, %2\n\t"
               "global_load_async_to_lds_b128 %3, %4, %2"
               :: "v"(ldsA), "v"(voff0), "s"(base),
                  "v"(ldsA + ldsStep), "v"(voff1)
               : "memory");

  const int t = tgt[(size_t)n * S_ + s0 + tid];
  float cdf = 0.0f, acc = 0.0f;

  for (int k = 0; k < NCHUNK; ++k) {
    if (k + 1 < NCHUNK) {
      // Prefetch chunk k+1 into the other buffer, then retire chunk k:
      // async loads complete in order, so cnt<=2 means chunk k has landed.
      const float*   p = base + (size_t)(k + 1) * CH * S_;
      const unsigned l = ((k + 1) & 1) ? ldsB : ldsA;
      asm volatile("global_load_async_to_lds_b128 %0, %1, %2\n\t"
                   "global_load_async_to_lds_b128 %3, %4, %2"
                   :: "v"(l), "v"(voff0), "s"(p),
                      "v"(l + ldsStep), "v"(voff1)
                   : "memory");
      asm volatile("s_wait_asynccnt 0x2" ::: "memory");
    } else {
      asm volatile("s_wait_asynccnt 0x0" ::: "memory");
    }
    __syncthreads();  // chunk k visible to all waves

    const float* lp = &lbuf[k & 1][tid];
#pragma unroll
    for (int r = 0; r < CH; ++r) {
      cdf += lp[r * TS];                       // inclusive cumsum over classes
      const int   c   = k * CH + r;
      const float ind = (c >= t) ? 1.0f : 0.0f;
      const float d   = cdf - ind;
      acc = fmaf(d, d, acc);
    }
    __syncthreads();  // buffer k&1 free before it is overwritten next iter
  }

  // Deterministic block tree-reduction (reuse lbuf; barrier above protects it)
  float* red = &lbuf[0][0];
  red[tid] = acc;
  __syncthreads();
  for (int sred = TS / 2; sred > 0; sred >>= 1) {
    if (tid < sred) red[tid] += red[tid + sred];
    __syncthreads();
  }
  if (tid == 0) partial[bid] = red[0];
}

// ---------------------------------------------------------------------------
// Final reduction: 512 partials summed with 8 chained V_WMMA_F32_16X16X4_F32
// against an all-ones B (D[m][n] = sum_k A[m][k] -> layout-independent total).
// One full wave32, EXEC all ones, fp32 accumulation, fixed order.
// ---------------------------------------------------------------------------
__global__ __launch_bounds__(32) void emd_reduce_wmma(const float* __restrict__ partial,
                                                      float* __restrict__ out) {
  const int lane = threadIdx.x;
  v8f d = {0.f, 0.f, 0.f, 0.f, 0.f, 0.f, 0.f, 0.f};
  const v2f bones = {1.0f, 1.0f};
#pragma unroll
  for (int i = 0; i < 8; ++i) {
    v2f a;
    a.x = partial[lane * 16 + 2 * i];
    a.y = partial[lane * 16 + 2 * i + 1];
    d = __builtin_amdgcn_wmma_f32_16x16x4_f32(
        /*neg_a=*/false, a, /*neg_b=*/false, bones,
        /*c_mod=*/(short)0, d, /*reuse_a=*/false, /*reuse_b=*/false);
  }
  // lanes 0..15 hold sum of D rows 0..7, lanes 16..31 rows 8..15
  float s = d[0] + d[1] + d[2] + d[3] + d[4] + d[5] + d[6] + d[7];
  __shared__ float tmp[32];
  tmp[lane] = s;
  __syncthreads();
  if (lane == 0) {
    // scale = 1/(S * N * S) = 2^-29 exactly
    out[0] = (tmp[0] + tmp[16]) * (1.0f / 536870912.0f);
  }
}

extern "C" void kernel_launch(void* const* d_in, const int* in_sizes, int n_in,
                              void* d_out, int out_size, void* d_ws, size_t ws_size,
                              hipStream_t stream) {
  const float* inp     = (const float*)d_in[0];
  const int*   tgt     = (const int*)d_in[1];
  float*       partial = (float*)d_ws;   // 512 floats of scratch

  emd_main<<<dim3(NBLK), dim3(TS), 0, stream>>>(inp, tgt, partial);
  emd_reduce_wmma<<<dim3(1), dim3(32), 0, stream>>>(partial, (float*)d_out);
}